// Ensemble_postprocess_triple_NMS_76656576299745
// MI455X (gfx1250) — compile-verified
//
#include <hip/hip_runtime.h>
#include <stdint.h>
#include <stddef.h>

#define NY 8400
#define NR 300
#define NF (NY + NR)
#define PY 16384
#define PR 512
#define PF 16384
#define WY ((NY + 31) / 32)   // 263
#define WR ((NR + 31) / 32)   // 10
#define WF ((NF + 31) / 32)   // 272
#define SCORE_THR 0.5f
#define IOU_THR 0.5f

typedef unsigned int u32x4 __attribute__((ext_vector_type(4)));
typedef int          i32x4 __attribute__((ext_vector_type(4)));
typedef int          i32x8 __attribute__((ext_vector_type(8)));

// ---------------------------------------------------------------- utilities
__device__ __forceinline__ unsigned sortable_f32(float f) {
    unsigned b = __float_as_uint(f);
    return (b & 0x80000000u) ? ~b : (b | 0x80000000u);
}

__device__ __forceinline__ unsigned lds_offset(const void* p) {
    // LDS generic address = {SHARED_BASE[63:32], lds_byte_offset[31:0]}
    return (unsigned)(uintptr_t)p;
}

// TDM: copy `nelem` contiguous fp32 (nelem*4 bytes) from global to LDS.
// Descriptor packing per CDNA5 ISA D# group layouts (1-D tile).
__device__ __forceinline__ void tdm_load_row_f32(unsigned lds_addr,
                                                 const void* gptr, int nelem) {
    unsigned long long ga = (unsigned long long)(uintptr_t)gptr;
    u32x4 g0;
    g0.x = 1u;                                        // count=1, user mode
    g0.y = lds_addr;                                  // lds_addr [63:32]
    g0.z = (unsigned)(ga & 0xFFFFFFFFull);            // global_addr [95:64]
    g0.w = (unsigned)((ga >> 32) & 0x1FFFFFFull) | (2u << 30);  // addr hi + type=2
    i32x8 g1;
    g1[0] = (2 << 16);                                // data_size=2 (4B), wg_mask=0
    g1[1] = (nelem & 0xFFFF) << 16;                   // tensor_dim0[15:0]
    g1[2] = ((nelem >> 16) & 0xFFFF) | (1 << 16);     // tensor_dim0 hi | tensor_dim1=1
    g1[3] = (nelem & 0xFFFF) << 16;                   // tile_dim0
    g1[4] = 1;                                        // tile_dim1=1, tile_dim2=0
    g1[5] = nelem;                                    // tensor_dim0_stride lo
    g1[6] = 0;
    g1[7] = 0;
#if __clang_major__ >= 23
    __builtin_amdgcn_tensor_load_to_lds(g0, g1, (i32x4)0, (i32x4)0, (i32x8)0, 0);
#else
    __builtin_amdgcn_tensor_load_to_lds(g0, g1, (i32x4)0, (i32x4)0, 0);
#endif
}

// ------------------------------------------------------------ prep kernels
__global__ void rt_max_kernel(const float* __restrict__ rt, float* __restrict__ rtMax) {
    __shared__ float red[512];
    int tid = threadIdx.x;
    float m = -1e30f;
    for (int n = tid; n < NR; n += 512) m = fmaxf(m, rt[n * 5 + 4]);
    red[tid] = m;
    __syncthreads();
    for (int s = 256; s > 0; s >>= 1) {
        if (tid < s) red[tid] = fmaxf(red[tid], red[tid + s]);
        __syncthreads();
    }
    if (tid == 0) rtMax[0] = red[0];
}

__global__ void prep_kernel(const float* __restrict__ yolo,   // [5][NY]
                            const float* __restrict__ rt,     // [NR][5]
                            const float* __restrict__ rtMax,
                            float4* __restrict__ yxy, float* __restrict__ yconf,
                            float4* __restrict__ rxy, float* __restrict__ rconf) {
    int i = blockIdx.x * blockDim.x + threadIdx.x;
    if (i < NY) {
        float cx = yolo[i], cy = yolo[NY + i];
        float w = yolo[2 * NY + i], h = yolo[3 * NY + i];
        float c = yolo[4 * NY + i];
        yxy[i] = make_float4(cx - 0.5f * w, cy - 0.5f * h, cx + 0.5f * w, cy + 0.5f * h);
        yconf[i] = c;
    }
    if (i < NR) {
        float cx = rt[i * 5 + 0], cy = rt[i * 5 + 1];
        float w = rt[i * 5 + 2], h = rt[i * 5 + 3];
        float c = rt[i * 5 + 4] / rtMax[0];
        rxy[i] = make_float4(cx - 0.5f * w, cy - 0.5f * h, cx + 0.5f * w, cy + 0.5f * h);
        rconf[i] = c;
    }
}

// ------------------------------------------------------------- sort kernels
// key = (~sortable(conf') << 32) | idx ; conf' = valid ? conf : -1
// ascending u64 sort == descending conf, ties broken by ascending idx (stable)
__global__ void pack_kernel(const float* __restrict__ conf, int N, int P,
                            unsigned long long* __restrict__ key, int* __restrict__ vc) {
    int i = blockIdx.x * blockDim.x + threadIdx.x;
    if (i == 0) *vc = 0;
    if (i >= P) return;
    if (i < N) {
        float c = conf[i];
        float k = (c >= SCORE_THR) ? c : -1.0f;
        unsigned s = sortable_f32(k);
        key[i] = ((unsigned long long)(~s) << 32) | (unsigned)i;
    } else {
        key[i] = ~0ull;
    }
}

__global__ void bitonic_global(unsigned long long* __restrict__ d, int k, int j) {
    int i = blockIdx.x * blockDim.x + threadIdx.x;
    int ixj = i ^ j;
    if (ixj > i) {
        unsigned long long a = d[i], b = d[ixj];
        bool up = ((i & k) == 0);
        if (up ? (a > b) : (a < b)) { d[i] = b; d[ixj] = a; }
    }
}

// handles all steps with j0 <= 256 inside one 512-element LDS tile
__global__ void bitonic_local(unsigned long long* __restrict__ d, int k, int j0) {
    __shared__ __align__(16) unsigned long long s[512];
    int base = blockIdx.x * 512;
    int t = threadIdx.x;
    int t2 = 2 * t;
    // async global -> LDS tile load (16B per lane, all lanes active)
    {
        unsigned lds = lds_offset(&s[t2]);
        asm volatile("global_load_async_to_lds_b128 %0, %1, off"
                     :: "v"(lds), "v"((unsigned long long)(uintptr_t)&d[base + t2])
                     : "memory");
        asm volatile("s_wait_asynccnt 0" ::: "memory");
    }
    __syncthreads();
    for (int j = j0; j > 0; j >>= 1) {
        int i = ((t & ~(j - 1)) << 1) | (t & (j - 1));
        int p = i + j;
        unsigned long long a = s[i], b = s[p];
        bool up = (((base + i) & k) == 0);
        if (up ? (a > b) : (a < b)) { s[i] = b; s[p] = a; }
        __syncthreads();
    }
    // async LDS -> global tile store
    {
        unsigned lds = lds_offset(&s[t2]);
        asm volatile("global_store_async_from_lds_b128 %0, %1, off"
                     :: "v"((unsigned long long)(uintptr_t)&d[base + t2]), "v"(lds)
                     : "memory");
        asm volatile("s_wait_asynccnt 0" ::: "memory");
    }
}

__global__ void gather_kernel(const unsigned long long* __restrict__ key,
                              const float4* __restrict__ boxes,
                              const float* __restrict__ conf, int N,
                              float4* __restrict__ sbox, float* __restrict__ sconf,
                              int* __restrict__ vc) {
    int i = blockIdx.x * blockDim.x + threadIdx.x;
    if (i >= N) return;
    unsigned idx = (unsigned)(key[i] & 0xFFFFFFFFu);
    float4 b = boxes[idx];
    float c = conf[idx];
    sbox[i] = b;
    sconf[i] = c;
    if (c >= SCORE_THR) atomicAdd(vc, 1);  // sorted => valid prefix length
}

// -------------------------------------------------------- suppression mask
// blockIdx.x = 32-column word, blockIdx.y = 256-row tile.
// Column boxes staged into LDS with one TDM tensor_load (512B strip).
__global__ void mask_kernel(const float4* __restrict__ sbox, const int* __restrict__ vcPtr,
                            unsigned* __restrict__ mask, int N, int W) {
    __shared__ __align__(16) float4 colbox[32];
    int vc = *vcPtr;
    int rowbase = blockIdx.y * blockDim.x;
    if (rowbase >= vc) return;          // uniform: whole tile past valid prefix

    int w = blockIdx.x;
    int j0 = w * 32;
    if (threadIdx.x < 32) {             // wave 0 issues the TDM op (EXEC ignored by TDM)
        tdm_load_row_f32(lds_offset(&colbox[0]), sbox + j0, 128);
        __builtin_amdgcn_s_wait_tensorcnt(0);
    }
    __syncthreads();

    int i = rowbase + threadIdx.x;
    if (i >= vc || i >= N) return;      // only rows that can ever be "kept" matter
    float4 a = sbox[i];
    float areaA = (a.z - a.x) * (a.w - a.y);
    unsigned bits = 0u;
#pragma unroll
    for (int c = 0; c < 32; ++c) {
        int j = j0 + c;
        float4 b = colbox[c];
        float lx = fmaxf(a.x, b.x), ly = fmaxf(a.y, b.y);
        float rx = fminf(a.z, b.z), ry = fminf(a.w, b.w);
        float iw = fmaxf(rx - lx, 0.0f), ih = fmaxf(ry - ly, 0.0f);
        float inter = iw * ih;
        float areaB = (b.z - b.x) * (b.w - b.y);
        float denom = areaA + areaB - inter;
        bool over = (denom > 0.0f) && (inter > IOU_THR * denom);  // iou > thr
        if (over && j > i && j < N) bits |= (1u << c);
    }
    mask[(size_t)i * W + w] = bits;
}

// ------------------------------------------------------ sequential NMS scan
__global__ void scan_kernel(const float4* __restrict__ sbox, const float* __restrict__ sconf,
                            const unsigned* __restrict__ mask, const int* __restrict__ vcPtr,
                            int N, int W, float* __restrict__ out) {
    __shared__ unsigned acc[288];
    int tid = threadIdx.x;
    for (int w = tid; w < W; w += blockDim.x) acc[w] = 0u;
    __syncthreads();
    int vc = *vcPtr;
    for (int i = 0; i < vc; ++i) {
        // bit i is only ever written by earlier *kept* iterations (each ends with a
        // barrier), so this read is stable and `keep` is uniform across the block.
        bool keep = ((acc[i >> 5] >> (i & 31)) & 1u) == 0u;
        if (keep) {
            for (int w = tid; w < W; w += blockDim.x) acc[w] |= mask[(size_t)i * W + w];
            __syncthreads();
        }
    }
    __syncthreads();
    for (int i = tid; i < N; i += blockDim.x) {
        bool keep = (i < vc) && (((acc[i >> 5] >> (i & 31)) & 1u) == 0u);
        float4 b = sbox[i];
        float s = sconf[i];
        out[(size_t)i * 5 + 0] = keep ? b.x : 0.0f;
        out[(size_t)i * 5 + 1] = keep ? b.y : 0.0f;
        out[(size_t)i * 5 + 2] = keep ? b.z : 0.0f;
        out[(size_t)i * 5 + 3] = keep ? b.w : 0.0f;
        out[(size_t)i * 5 + 4] = keep ? s : 0.0f;
    }
}

// -------------------------------------------------- stage-3 input assembly
__global__ void concat_kernel(const float* __restrict__ ybs, const float* __restrict__ rbs,
                              float4* __restrict__ fbox, float* __restrict__ fconf) {
    int i = blockIdx.x * blockDim.x + threadIdx.x;
    if (i >= NF) return;
    const float* src = (i < NY) ? (ybs + (size_t)i * 5) : (rbs + (size_t)(i - NY) * 5);
    fbox[i] = make_float4(src[0], src[1], src[2], src[3]);
    fconf[i] = src[4];
}

// ------------------------------------------------------------------- launch
extern "C" void kernel_launch(void* const* d_in, const int* in_sizes, int n_in,
                              void* d_out, int out_size, void* d_ws, size_t ws_size,
                              hipStream_t stream) {
    const float* yolo = (const float*)d_in[0];   // [1,5,8400]
    const float* rt   = (const float*)d_in[1];   // [1,300,5]
    float* out = (float*)d_out;                  // [8700,5] fp32

    // bump allocator over d_ws (256B aligned)
    char* p = (char*)d_ws;
    auto alloc = [&](size_t bytes) -> void* {
        void* r = (void*)p;
        p += (bytes + 255) & ~(size_t)255;
        return r;
    };
    float4* yxy   = (float4*)alloc(sizeof(float4) * NY);
    float*  yconf = (float*) alloc(sizeof(float) * NY);
    float4* rxy   = (float4*)alloc(sizeof(float4) * NR);
    float*  rconf = (float*) alloc(sizeof(float) * NR);
    float*  rtMax = (float*) alloc(sizeof(float));
    unsigned long long* keybuf = (unsigned long long*)alloc(sizeof(unsigned long long) * PY);
    float4* sbox  = (float4*)alloc(sizeof(float4) * (NF + 64));   // +64: TDM tile overrun pad
    float*  sconf = (float*) alloc(sizeof(float) * (NF + 64));
    int*    vc    = (int*)   alloc(sizeof(int));
    float*  ybs   = (float*) alloc(sizeof(float) * NY * 5);
    float*  rbs   = (float*) alloc(sizeof(float) * NR * 5);
    float4* fbox  = (float4*)alloc(sizeof(float4) * NF);
    float*  fconf = (float*) alloc(sizeof(float) * NF);
    unsigned* mask = (unsigned*)alloc(sizeof(unsigned) * (size_t)(NF + 64) * WF);

    auto sort_pass = [&](unsigned long long* key, int P) {
        int blocksG = P / 256;
        int blocksL = P / 512;
        for (int k = 2; k <= P; k <<= 1) {
            int j = k >> 1;
            for (; j >= 512; j >>= 1)
                bitonic_global<<<blocksG, 256, 0, stream>>>(key, k, j);
            bitonic_local<<<blocksL, 256, 0, stream>>>(key, k, j);  // j <= 256
        }
    };

    auto run_stage = [&](const float4* boxes, const float* conf, int N, int P, int W,
                         float* stage_out) {
        int pb = (P + 255) / 256;
        int nb = (N + 255) / 256;
        pack_kernel<<<pb, 256, 0, stream>>>(conf, N, P, keybuf, vc);
        sort_pass(keybuf, P);
        gather_kernel<<<nb, 256, 0, stream>>>(keybuf, boxes, conf, N, sbox, sconf, vc);
        mask_kernel<<<dim3(W, nb), 256, 0, stream>>>(sbox, vc, mask, N, W);
        scan_kernel<<<1, 512, 0, stream>>>(sbox, sconf, mask, vc, N, W, stage_out);
    };

    // preprocessing
    rt_max_kernel<<<1, 512, 0, stream>>>(rt, rtMax);
    prep_kernel<<<(NY + 255) / 256, 256, 0, stream>>>(yolo, rt, rtMax,
                                                      yxy, yconf, rxy, rconf);
    // stage 1: YOLO NMS
    run_stage(yxy, yconf, NY, PY, WY, ybs);
    // stage 2: RT-DETR NMS
    run_stage(rxy, rconf, NR, PR, WR, rbs);
    // stage 3: ensemble NMS
    concat_kernel<<<(NF + 255) / 256, 256, 0, stream>>>(ybs, rbs, fbox, fconf);
    run_stage(fbox, fconf, NF, PF, WF, out);

    (void)in_sizes; (void)n_in; (void)out_size; (void)ws_size;
}